// GGAT_14130442404358
// MI455X (gfx1250) — compile-verified
//
#include <hip/hip_runtime.h>
#include <hip/hip_bf16.h>
#include <cstddef>

// ---------------------------------------------------------------------------
// Problem constants (derived from in_sizes at launch, but fixed by reference):
//   N=50000 nodes, E=800000 edges, F_IN=128, C=32, H=4, H*C=128, 3C=96
// ---------------------------------------------------------------------------

typedef __attribute__((ext_vector_type(2))) float v2f;
typedef __attribute__((ext_vector_type(8))) float v8f;

#define FIN   128
#define HC    128
#define COUT  32
#define NHEAD 4
#define G3C   96

// ---- order-preserving float <-> uint encoding for atomic segment-max ------
__device__ __forceinline__ unsigned fOrd(float f) {
  unsigned u = __float_as_uint(f);
  return (u & 0x80000000u) ? ~u : (u | 0x80000000u);
}
__device__ __forceinline__ float fUnord(unsigned u) {
  return __uint_as_float((u & 0x80000000u) ? (u & 0x7fffffffu) : ~u);
}

// ---------------------------------------------------------------------------
// Zero-init of the contiguous accumulator region (amaxU | denom | gat).
// All-zero bits works for both the f32 accumulators and the ordered-uint max
// (0u is the minimum of the ordered encoding).
// ---------------------------------------------------------------------------
__global__ void init_zero_kernel(unsigned* __restrict__ p, int count) {
  int i = blockIdx.x * blockDim.x + threadIdx.x;
  if (i < count) p[i] = 0u;
}

// ---------------------------------------------------------------------------
// Full-precision WMMA GEMM: C[M x Ncols] = A[M x K] * B
//   TRANS_B == 0 : B is [K x Ncols] row-major          (x @ W)
//   TRANS_B == 1 : B is [Ncols x K] row-major (B^T)    (inp @ W_ih^T)
// One wave computes one 16x16 tile with V_WMMA_F32_16X16X4_F32 (K-step 4).
// Fragment layouts per CDNA5 ISA 7.12.2:
//   A: lane(r,hf) vgpr i = A[m0+r][k + 2*hf + i]      (contiguous f32 pair)
//   B: lane(r,hf) vgpr i = B[k + 2*hf + i][n0 + r]
//   D: vgpr j            = D[m0 + j + 8*hf][n0 + r]
// ---------------------------------------------------------------------------
template <int TRANS_B>
__global__ void wmma_gemm_f32_kernel(const float* __restrict__ A,
                                     const float* __restrict__ B,
                                     float* __restrict__ C,
                                     int M, int K, int Ncols) {
  const int lane = threadIdx.x & 31;
  const int wave = threadIdx.x >> 5;
  const int r    = lane & 15;
  const int hf   = lane >> 4;
  const int m0   = blockIdx.x * 16;
  const int n0   = (blockIdx.y * 4 + wave) * 16;
  if (m0 >= M || n0 >= Ncols) return;

  const float* arow = A + (size_t)(m0 + r) * K + 2 * hf;
  v8f acc = {};
  if (TRANS_B) {
    const float* brow = B + (size_t)(n0 + r) * K + 2 * hf;
#pragma unroll 8
    for (int k = 0; k < K; k += 4) {
      v2f a, b;
      a[0] = arow[k];     a[1] = arow[k + 1];
      b[0] = brow[k];     b[1] = brow[k + 1];
      acc = __builtin_amdgcn_wmma_f32_16x16x4_f32(false, a, false, b,
                                                  (short)0, acc, false, false);
    }
  } else {
#pragma unroll 8
    for (int k = 0; k < K; k += 4) {
      v2f a, b;
      a[0] = arow[k];     a[1] = arow[k + 1];
      b[0] = B[(size_t)(k + 2 * hf)     * Ncols + n0 + r];
      b[1] = B[(size_t)(k + 2 * hf + 1) * Ncols + n0 + r];
      acc = __builtin_amdgcn_wmma_f32_16x16x4_f32(false, a, false, b,
                                                  (short)0, acc, false, false);
    }
  }
#pragma unroll
  for (int j = 0; j < 8; ++j)
    C[(size_t)(m0 + j + 8 * hf) * Ncols + n0 + r] = acc[j];
}

// ---------------------------------------------------------------------------
// a_src[n,h] = dot(xw[n, h*32 : h*32+32], att_src[h]);  same for a_dst.
// One wave32 per (node, head); lane == channel; 5-step shuffle reduction.
// ---------------------------------------------------------------------------
__global__ void attn_scores_kernel(const float* __restrict__ xw,
                                   const float* __restrict__ att_src,
                                   const float* __restrict__ att_dst,
                                   float* __restrict__ a_src,
                                   float* __restrict__ a_dst, int N) {
  int gw   = (blockIdx.x * blockDim.x + threadIdx.x) >> 5;
  int lane = threadIdx.x & 31;
  int n = gw >> 2, h = gw & 3;
  if (n >= N) return;
  float xv = xw[(size_t)n * HC + h * COUT + lane];
  float ps = xv * att_src[h * COUT + lane];
  float pd = xv * att_dst[h * COUT + lane];
#pragma unroll
  for (int off = 16; off > 0; off >>= 1) {
    ps += __shfl_down(ps, off, 32);
    pd += __shfl_down(pd, off, 32);
  }
  if (lane == 0) {
    a_src[n * NHEAD + h] = ps;
    a_dst[n * NHEAD + h] = pd;
  }
}

// ---------------------------------------------------------------------------
// Edge pass 1: alpha = leaky_relu(a_src[src]+a_dst[dst]); segment-max via
// integer atomic max on the order-preserving encoding.
// ---------------------------------------------------------------------------
__global__ void edge_logits_max_kernel(const long long* __restrict__ ei,
                                       const float* __restrict__ a_src,
                                       const float* __restrict__ a_dst,
                                       float* __restrict__ alphaBuf,
                                       unsigned* __restrict__ amaxU, int E) {
  int e = blockIdx.x * blockDim.x + threadIdx.x;
  if (e >= E) return;
  int s = (int)ei[e];
  int d = (int)ei[(size_t)E + e];
#pragma unroll
  for (int h = 0; h < NHEAD; ++h) {
    float a = a_src[s * NHEAD + h] + a_dst[d * NHEAD + h];
    a = a > 0.f ? a : 0.2f * a;               // leaky_relu, slope 0.2
    alphaBuf[(size_t)e * NHEAD + h] = a;
    atomicMax(&amaxU[d * NHEAD + h], fOrd(a));
  }
}

// ---------------------------------------------------------------------------
// Edge pass 2: ex = exp(alpha - amax[dst]); denom[dst] += ex (overwrite alpha)
// ---------------------------------------------------------------------------
__global__ void edge_exp_sum_kernel(const long long* __restrict__ ei,
                                    float* __restrict__ alphaBuf,
                                    const unsigned* __restrict__ amaxU,
                                    float* __restrict__ denom, int E) {
  int e = blockIdx.x * blockDim.x + threadIdx.x;
  if (e >= E) return;
  int d = (int)ei[(size_t)E + e];
#pragma unroll
  for (int h = 0; h < NHEAD; ++h) {
    float ex = __expf(alphaBuf[(size_t)e * NHEAD + h] - fUnord(amaxU[d * NHEAD + h]));
    alphaBuf[(size_t)e * NHEAD + h] = ex;
    atomicAdd(&denom[d * NHEAD + h], ex);
  }
}

// ---------------------------------------------------------------------------
// Edge pass 3: gat[dst, h, c] += xw[src, h, c] * ex / (denom[dst,h] + 1e-16).
// 128 lanes = one full (H*C) channel vector per edge; EPB edges per block.
// With the whole working set resident in the 192 MB L2, these random 512 B
// gathers + f32 atomic adds run at L2 rather than HBM-random speed.
// ---------------------------------------------------------------------------
#define EPB 8
__global__ void edge_scatter_kernel(const long long* __restrict__ ei,
                                    const float* __restrict__ alphaBuf,
                                    const float* __restrict__ denom,
                                    const float* __restrict__ xw,
                                    float* __restrict__ gat, int E) {
  int lane = threadIdx.x;           // 0..127
  int h    = lane >> 5;             // head
  int e0   = blockIdx.x * EPB;
  int eEnd = e0 + EPB; if (eEnd > E) eEnd = E;
  for (int e = e0; e < eEnd; ++e) {
    int s = (int)ei[e];
    int d = (int)ei[(size_t)E + e];
    float ex   = alphaBuf[(size_t)e * NHEAD + h];
    float attn = ex / (denom[d * NHEAD + h] + 1e-16f);
    atomicAdd(&gat[(size_t)d * HC + lane], xw[(size_t)s * HC + lane] * attn);
  }
}

// ---------------------------------------------------------------------------
// In-place bias + ELU on the aggregated GAT output.
// ---------------------------------------------------------------------------
__global__ void bias_elu_kernel(float* __restrict__ gat,
                                const float* __restrict__ bias, int total) {
  int i = blockIdx.x * blockDim.x + threadIdx.x;
  if (i >= total) return;
  float v = gat[i] + bias[i & (HC - 1)];
  gat[i] = v > 0.f ? v : (__expf(v) - 1.f);
}

// ---------------------------------------------------------------------------
// GRU cell. gi[N x 96] already computed by WMMA GEMM (no bias).
// One wave32 per node: lane = output channel. W_hh (96x32) staged in LDS;
// h-row broadcast via __shfl (wave32 width).
// ---------------------------------------------------------------------------
__global__ void gru_kernel(const float* __restrict__ gi,
                           const float* __restrict__ hstate,
                           const float* __restrict__ W_hh,
                           const float* __restrict__ b_ih,
                           const float* __restrict__ b_hh,
                           float* __restrict__ out, int N) {
  __shared__ float sW[G3C * COUT];          // 12 KB
  for (int i = threadIdx.x; i < G3C * COUT; i += blockDim.x) sW[i] = W_hh[i];
  __syncthreads();

  int lane = threadIdx.x & 31;
  int wid  = threadIdx.x >> 5;
  int n    = blockIdx.x * (blockDim.x >> 5) + wid;
  if (n >= N) return;

  float hv = hstate[(size_t)n * COUT + lane];
  float hr = 0.f, hz = 0.f, hn = 0.f;
#pragma unroll
  for (int k = 0; k < COUT; ++k) {
    float hk = __shfl(hv, k, 32);
    hr += hk * sW[lane * COUT + k];
    hz += hk * sW[(COUT + lane) * COUT + k];
    hn += hk * sW[(2 * COUT + lane) * COUT + k];
  }
  float i_r = gi[(size_t)n * G3C + lane]            + b_ih[lane];
  float i_z = gi[(size_t)n * G3C + COUT + lane]     + b_ih[COUT + lane];
  float i_n = gi[(size_t)n * G3C + 2 * COUT + lane] + b_ih[2 * COUT + lane];
  float h_r = hr + b_hh[lane];
  float h_z = hz + b_hh[COUT + lane];
  float h_n = hn + b_hh[2 * COUT + lane];

  float r  = 1.f / (1.f + __expf(-(i_r + h_r)));
  float z  = 1.f / (1.f + __expf(-(i_z + h_z)));
  float nn = tanhf(i_n + r * h_n);
  out[(size_t)n * COUT + lane] = (1.f - z) * nn + z * hv;
}

// ---------------------------------------------------------------------------
// Host-side orchestration (graph-capture safe: kernels only, all on stream).
// ---------------------------------------------------------------------------
extern "C" void kernel_launch(void* const* d_in, const int* in_sizes, int n_in,
                              void* d_out, int out_size, void* d_ws, size_t ws_size,
                              hipStream_t stream) {
  const float*     x       = (const float*)d_in[0];
  const long long* ei      = (const long long*)d_in[1];   // int64 edge_index [2,E]
  /* d_in[2] = edge_attr (ignored, edge_dim=None) */
  const float*     h       = (const float*)d_in[3];
  const float*     W       = (const float*)d_in[4];
  const float*     att_src = (const float*)d_in[5];
  const float*     att_dst = (const float*)d_in[6];
  const float*     bias    = (const float*)d_in[7];
  const float*     W_ih    = (const float*)d_in[8];
  const float*     W_hh    = (const float*)d_in[9];
  const float*     b_ih    = (const float*)d_in[10];
  const float*     b_hh    = (const float*)d_in[11];
  float*           out     = (float*)d_out;

  const int N = in_sizes[0] / FIN;   // 50000
  const int E = in_sizes[1] / 2;     // 800000

  // ---- workspace layout (all 4-byte words) ----
  // xw (N*128) | a_src (N*4) | a_dst (N*4) |
  // [ amaxU (N*4) | denom (N*4) | gat (N*128) ]  <- contiguous zero region
  // alphaBuf (E*4) | gi (N*96)
  float*    xw       = (float*)d_ws;
  float*    a_src    = xw + (size_t)N * HC;
  float*    a_dst    = a_src + (size_t)N * NHEAD;
  unsigned* amaxU    = (unsigned*)(a_dst + (size_t)N * NHEAD);
  float*    denom    = (float*)(amaxU + (size_t)N * NHEAD);
  float*    gat      = denom + (size_t)N * NHEAD;
  float*    alphaBuf = gat + (size_t)N * HC;
  float*    gi       = alphaBuf + (size_t)E * NHEAD;

  // 1) zero accumulators (amaxU | denom | gat contiguous)
  {
    int zeroCount = N * (NHEAD + NHEAD + HC);
    init_zero_kernel<<<(zeroCount + 255) / 256, 256, 0, stream>>>(amaxU, zeroCount);
  }
  // 2) xw = x @ W   [N,128] = [N,128]x[128,128]  (WMMA f32 16x16x4)
  wmma_gemm_f32_kernel<0><<<dim3(N / 16, 2), 128, 0, stream>>>(x, W, xw, N, FIN, HC);
  // 3) attention logits per (node, head)
  attn_scores_kernel<<<(N * NHEAD + 7) / 8, 256, 0, stream>>>(xw, att_src, att_dst,
                                                              a_src, a_dst, N);
  // 4) edge logits + segment max
  edge_logits_max_kernel<<<(E + 255) / 256, 256, 0, stream>>>(ei, a_src, a_dst,
                                                              alphaBuf, amaxU, E);
  // 5) exp + segment sum
  edge_exp_sum_kernel<<<(E + 255) / 256, 256, 0, stream>>>(ei, alphaBuf, amaxU, denom, E);
  // 6) message scatter
  edge_scatter_kernel<<<(E + EPB - 1) / EPB, HC, 0, stream>>>(ei, alphaBuf, denom,
                                                              xw, gat, E);
  // 7) bias + ELU (in place -> GRU input)
  bias_elu_kernel<<<(N * HC + 255) / 256, 256, 0, stream>>>(gat, bias, N * HC);
  // 8) gi = elu_out @ W_ih^T   [N,96] = [N,128]x[128,96]  (WMMA, transposed B)
  wmma_gemm_f32_kernel<1><<<dim3(N / 16, 2), 128, 0, stream>>>(gat, W_ih, gi, N, HC, G3C);
  // 9) GRU gates -> d_out [N,32]
  gru_kernel<<<(N + 7) / 8, 256, 0, stream>>>(gi, h, W_hh, b_ih, b_hh, out, N);
  (void)n_in; (void)out_size; (void)ws_size;
}